// GINE_47485158425273
// MI455X (gfx1250) — compile-verified
//
#include <hip/hip_runtime.h>

typedef float v2f __attribute__((ext_vector_type(2)));
typedef float v8f __attribute__((ext_vector_type(8)));

__device__ __forceinline__ v8f wmma4(v2f a, v2f b, v8f c) {
    // D(16x16,f32) = A(16x4,f32) x B(4x16,f32) + C
    return __builtin_amdgcn_wmma_f32_16x16x4_f32(false, a, false, b, (short)0, c,
                                                 false, false);
}

// ---------------------------------------------------------------- utilities
__global__ void zero_k(float* __restrict__ p, size_t n) {
    size_t i = (size_t)blockIdx.x * blockDim.x + threadIdx.x;
    if (i < n) p[i] = 0.0f;
}

// ---------------------------------------------------- layer 1 edge (F = 2)
__global__ void edge_l1(const float* __restrict__ x, const float* __restrict__ ea,
                        const int* __restrict__ src, const int* __restrict__ dst,
                        const float* __restrict__ w, const float* __restrict__ b,
                        float* __restrict__ agg, int E) {
    int e = blockIdx.x * blockDim.x + threadIdx.x;
    if (e >= E) return;
    float a0 = b[0], a1 = b[1];
    const float* ep = ea + (size_t)e * 7;
#pragma unroll
    for (int k = 0; k < 7; ++k) {
        float v = ep[k];
        a0 += v * w[k * 2 + 0];
        a1 += v * w[k * 2 + 1];
    }
    int s = src[e], d = dst[e];
    float m0 = x[(size_t)s * 2 + 0] + a0; m0 = m0 > 0.f ? m0 : 0.f;
    float m1 = x[(size_t)s * 2 + 1] + a1; m1 = m1 > 0.f ? m1 : 0.f;
    atomicAdd(&agg[(size_t)d * 2 + 0], m0);
    atomicAdd(&agg[(size_t)d * 2 + 1], m1);
}

// ------------------------------------------------- layers 2/3 edge (H = 64)
// One wave per edge: 32 lanes x float2 = coalesced 256B row gather; edge-attr
// broadcast via wave shuffles (no LDS, no barrier). Atomic scatter-add.
__global__ __launch_bounds__(256) void edge_lH(
        const float* __restrict__ xin, const float* __restrict__ ea,
        const int* __restrict__ src, const int* __restrict__ dst,
        const float* __restrict__ w, const float* __restrict__ b,
        float* __restrict__ agg, int E) {
    int lane = threadIdx.x & 31;
    int wv   = threadIdx.x >> 5;
    int e = blockIdx.x * 8 + wv;
    if (e >= E) return;                      // wave-uniform exit
    float eav = (lane < 7) ? ea[(size_t)e * 7 + lane] : 0.f;
    int s = src[e], d = dst[e];
    const float2* wp = (const float2*)w;     // [7][32] float2 view of [7][64]
    float2 acc = ((const float2*)b)[lane];
#pragma unroll
    for (int k = 0; k < 7; ++k) {
        float evk = __shfl(eav, k, 32);
        float2 wk = wp[k * 32 + lane];
        acc.x += evk * wk.x;
        acc.y += evk * wk.y;
    }
    float2 xv = ((const float2*)(xin + (size_t)s * 64))[lane];
    float m0 = xv.x + acc.x; m0 = m0 > 0.f ? m0 : 0.f;
    float m1 = xv.y + acc.y; m1 = m1 > 0.f ? m1 : 0.f;
    float* ap = agg + (size_t)d * 64 + 2 * lane;
    atomicAdd(ap + 0, m0);
    atomicAdd(ap + 1, m1);
}

// --------------------------------------------- layer 1 GEMM1: [N,2]x[2,64]
// K padded 2->4: lanes 16..31 carry K=2,3 which are zeroed.
__global__ __launch_bounds__(256) void gemm1_l1(
        const float* __restrict__ x, const float* __restrict__ agg,
        const float* __restrict__ eps, const float* __restrict__ w1,
        const float* __restrict__ b1, float* __restrict__ zb,
        float* __restrict__ sums, float* __restrict__ sumsq, int N) {
    int lane = threadIdx.x & 31;
    int wid  = threadIdx.x >> 5;
    int rowBase = (blockIdx.x * 8 + wid) * 16;
    int r  = rowBase + (lane & 15);
    int rr = r < N ? r : (N - 1);
    bool fullTile = (rowBase + 16 <= N);
    bool lowK = (lane < 16);
    float ep = 1.0f + eps[0];
    v2f a;
    if (lowK) {
        a.x = ep * x[(size_t)rr * 2 + 0] + agg[(size_t)rr * 2 + 0];
        a.y = ep * x[(size_t)rr * 2 + 1] + agg[(size_t)rr * 2 + 1];
    } else { a.x = 0.f; a.y = 0.f; }
    int cbase = lane & 15;
    int moff  = (lane >> 4) * 8;
#pragma unroll
    for (int nt = 0; nt < 4; ++nt) {
        int col = nt * 16 + cbase;
        v2f bf;
        if (lowK) { bf.x = w1[col]; bf.y = w1[64 + col]; }
        else      { bf.x = 0.f;     bf.y = 0.f; }
        v8f acc = {0.f,0.f,0.f,0.f,0.f,0.f,0.f,0.f};
        acc = wmma4(a, bf, acc);
        float bb = b1[col];
        float ls = 0.f, lq = 0.f;
        float* zp = zb + (size_t)(rowBase + moff) * 64 + col;
        if (fullTile) {
#pragma unroll
            for (int jj = 0; jj < 8; ++jj) {
                float z = acc[jj] + bb;
                zp[(size_t)jj * 64] = z; ls += z; lq += z * z;
            }
        } else {
#pragma unroll
            for (int jj = 0; jj < 8; ++jj) {
                float z = acc[jj] + bb;
                if (rowBase + moff + jj < N) { zp[(size_t)jj * 64] = z; ls += z; lq += z * z; }
            }
        }
        atomicAdd(&sums[col], ls);
        atomicAdd(&sumsq[col], lq);
    }
}

// ------------------------------------- GEMM1 (H): z = ((1+eps)x + agg) @ w1
__global__ __launch_bounds__(256) void gemm1_h(
        const float* __restrict__ xin, const float* __restrict__ agg,
        const float* __restrict__ eps, const float* __restrict__ w1,
        const float* __restrict__ b1, float* __restrict__ zb,
        float* __restrict__ sums, float* __restrict__ sumsq, int N) {
    __shared__ float sB[64 * 64];
    for (int i = threadIdx.x; i < 1024; i += 256)
        ((float4*)sB)[i] = ((const float4*)w1)[i];
    __syncthreads();

    int lane = threadIdx.x & 31;
    int wid  = threadIdx.x >> 5;
    int rowBase = (blockIdx.x * 8 + wid) * 16;
    int r  = rowBase + (lane & 15);
    int rr = r < N ? r : (N - 1);
    bool fullTile = (rowBase + 16 <= N);
    int khalf = (lane >> 4) * 2;
    float ep = 1.0f + eps[0];

    const float2* xi = (const float2*)(xin + (size_t)rr * 64);
    const float2* ag = (const float2*)(agg + (size_t)rr * 64);
    v2f A[16];
#pragma unroll
    for (int s = 0; s < 16; ++s) {
        int k2 = (4 * s + khalf) >> 1;
        float2 xv = xi[k2];
        float2 av = ag[k2];
        A[s].x = ep * xv.x + av.x;
        A[s].y = ep * xv.y + av.y;
    }
    int cbase = lane & 15;
    int moff  = (lane >> 4) * 8;
#pragma unroll
    for (int nt = 0; nt < 4; ++nt) {
        int col = nt * 16 + cbase;
        v8f acc = {0.f,0.f,0.f,0.f,0.f,0.f,0.f,0.f};
#pragma unroll
        for (int s = 0; s < 16; ++s) {
            int k = 4 * s + khalf;
            v2f bf;
            bf.x = sB[k * 64 + col];
            bf.y = sB[(k + 1) * 64 + col];
            acc = wmma4(A[s], bf, acc);
        }
        float bb = b1[col];
        float ls = 0.f, lq = 0.f;
        float* zp = zb + (size_t)(rowBase + moff) * 64 + col;
        if (fullTile) {
#pragma unroll
            for (int jj = 0; jj < 8; ++jj) {
                float z = acc[jj] + bb;
                zp[(size_t)jj * 64] = z; ls += z; lq += z * z;
            }
        } else {
#pragma unroll
            for (int jj = 0; jj < 8; ++jj) {
                float z = acc[jj] + bb;
                if (rowBase + moff + jj < N) { zp[(size_t)jj * 64] = z; ls += z; lq += z * z; }
            }
        }
        atomicAdd(&sums[col], ls);
        atomicAdd(&sumsq[col], lq);
    }
}

// ----------------------------------------------------------- BN finalize
__global__ void bn_fin(const float* __restrict__ sums, const float* __restrict__ sumsq,
                       const float* __restrict__ g, const float* __restrict__ bt,
                       float* __restrict__ scale, float* __restrict__ shift, int N) {
    int j = threadIdx.x;
    if (j < 64) {
        float inv = 1.0f / (float)N;
        float m = sums[j] * inv;
        float v = sumsq[j] * inv - m * m;
        float sc = g[j] * rsqrtf(v + 1e-5f);
        scale[j] = sc;
        shift[j] = bt[j] - m * sc;
    }
}

// ------------------------- GEMM2: out = relu( relu(bn(z)) @ w2 + b2 )
__global__ __launch_bounds__(256) void gemm2_h(
        const float* __restrict__ zb, const float* __restrict__ scale,
        const float* __restrict__ shift, const float* __restrict__ w2,
        const float* __restrict__ b2, float* __restrict__ out, int N) {
    __shared__ float sB[64 * 64];
    __shared__ float ssc[64];
    __shared__ float ssh[64];
    for (int i = threadIdx.x; i < 1024; i += 256)
        ((float4*)sB)[i] = ((const float4*)w2)[i];
    if (threadIdx.x < 64) { ssc[threadIdx.x] = scale[threadIdx.x];
                            ssh[threadIdx.x] = shift[threadIdx.x]; }
    __syncthreads();

    int lane = threadIdx.x & 31;
    int wid  = threadIdx.x >> 5;
    int rowBase = (blockIdx.x * 8 + wid) * 16;
    int r  = rowBase + (lane & 15);
    int rr = r < N ? r : (N - 1);
    bool fullTile = (rowBase + 16 <= N);
    int khalf = (lane >> 4) * 2;

    const float2* zi = (const float2*)(zb + (size_t)rr * 64);
    v2f A[16];
#pragma unroll
    for (int s = 0; s < 16; ++s) {
        int k = 4 * s + khalf;
        float2 zv = zi[k >> 1];
        float a0 = zv.x * ssc[k] + ssh[k];
        float a1 = zv.y * ssc[k + 1] + ssh[k + 1];
        A[s].x = a0 > 0.f ? a0 : 0.f;
        A[s].y = a1 > 0.f ? a1 : 0.f;
    }
    int cbase = lane & 15;
    int moff  = (lane >> 4) * 8;
#pragma unroll
    for (int nt = 0; nt < 4; ++nt) {
        int col = nt * 16 + cbase;
        v8f acc = {0.f,0.f,0.f,0.f,0.f,0.f,0.f,0.f};
#pragma unroll
        for (int s = 0; s < 16; ++s) {
            int k = 4 * s + khalf;
            v2f bf;
            bf.x = sB[k * 64 + col];
            bf.y = sB[(k + 1) * 64 + col];
            acc = wmma4(A[s], bf, acc);
        }
        float bb = b2[col];
        float* op = out + (size_t)(rowBase + moff) * 64 + col;
        if (fullTile) {
#pragma unroll
            for (int jj = 0; jj < 8; ++jj) {
                float o = acc[jj] + bb;
                op[(size_t)jj * 64] = o > 0.f ? o : 0.f;
            }
        } else {
#pragma unroll
            for (int jj = 0; jj < 8; ++jj) {
                float o = acc[jj] + bb;
                if (rowBase + moff + jj < N) op[(size_t)jj * 64] = o > 0.f ? o : 0.f;
            }
        }
    }
}

// ------------ fused head: out[n] = leaky(feat@reg_w + reg_b) @ end_w + end_b
__global__ __launch_bounds__(256) void reg_end(
        const float* __restrict__ xin, const float* __restrict__ rw,
        const float* __restrict__ rb, const float* __restrict__ ew,
        const float* __restrict__ eb, float* __restrict__ out, int N, int R) {
    int lane = threadIdx.x & 31;
    int wid  = threadIdx.x >> 5;
    int rowBase = (blockIdx.x * 8 + wid) * 16;
    int r  = rowBase + (lane & 15);
    int rr = r < N ? r : (N - 1);
    int khalf = (lane >> 4) * 2;

    const float2* xi = (const float2*)(xin + (size_t)rr * 64);
    v2f A[16];
#pragma unroll
    for (int s = 0; s < 16; ++s) {
        float2 v = xi[(4 * s + khalf) >> 1];
        A[s].x = v.x; A[s].y = v.y;
    }
    float accr[8] = {0.f,0.f,0.f,0.f,0.f,0.f,0.f,0.f};
    int cbase = lane & 15;
    int tiles = (R + 15) / 16;
    for (int t = 0; t < tiles; ++t) {
        int col = t * 16 + cbase;
        bool cv = col < R;
        int cc = cv ? col : 0;
        v8f c = {0.f,0.f,0.f,0.f,0.f,0.f,0.f,0.f};
#pragma unroll
        for (int s = 0; s < 16; ++s) {
            int k = 4 * s + khalf;
            v2f bf;
            bf.x = cv ? rw[k * R + cc]       : 0.f;
            bf.y = cv ? rw[(k + 1) * R + cc] : 0.f;
            c = wmma4(A[s], bf, c);
        }
        float rbv = cv ? rb[cc] : 0.f;
        float ewv = cv ? ew[cc] : 0.f;
#pragma unroll
        for (int jj = 0; jj < 8; ++jj) {
            float v = c[jj] + rbv;
            v = v >= 0.f ? v : 0.01f * v;   // LeakyReLU
            accr[jj] += v * ewv;
        }
    }
    // reduce over the 16 lanes of each half-wave (each lane holds one column)
#pragma unroll
    for (int jj = 0; jj < 8; ++jj) {
        float v = accr[jj];
        v += __shfl_xor(v, 1, 16);
        v += __shfl_xor(v, 2, 16);
        v += __shfl_xor(v, 4, 16);
        v += __shfl_xor(v, 8, 16);
        accr[jj] = v;
    }
    if ((lane & 15) == 0) {
        float ebv = eb[0];
        int moff = (lane >> 4) * 8;
#pragma unroll
        for (int jj = 0; jj < 8; ++jj) {
            int grow = rowBase + jj + moff;
            if (grow < N) out[grow] = accr[jj] + ebv;
        }
    }
}

// ---------------------------------------------------------------- driver
extern "C" void kernel_launch(void* const* d_in, const int* in_sizes, int n_in,
                              void* d_out, int out_size, void* d_ws, size_t ws_size,
                              hipStream_t stream) {
    (void)n_in; (void)out_size; (void)ws_size;
    const float* x     = (const float*)d_in[0];
    const float* ea    = (const float*)d_in[1];
    const int*   ei    = (const int*)d_in[2];
    const float* eps1  = (const float*)d_in[4];
    const float* el1_w = (const float*)d_in[5];
    const float* el1_b = (const float*)d_in[6];
    const float* n1_w1 = (const float*)d_in[7];
    const float* n1_b1 = (const float*)d_in[8];
    const float* n1_g  = (const float*)d_in[9];
    const float* n1_bt = (const float*)d_in[10];
    const float* n1_w2 = (const float*)d_in[11];
    const float* n1_b2 = (const float*)d_in[12];
    const float* eps2  = (const float*)d_in[13];
    const float* el2_w = (const float*)d_in[14];
    const float* el2_b = (const float*)d_in[15];
    const float* n2_w1 = (const float*)d_in[16];
    const float* n2_b1 = (const float*)d_in[17];
    const float* n2_g  = (const float*)d_in[18];
    const float* n2_bt = (const float*)d_in[19];
    const float* n2_w2 = (const float*)d_in[20];
    const float* n2_b2 = (const float*)d_in[21];
    const float* reg_w = (const float*)d_in[22];
    const float* reg_b = (const float*)d_in[23];
    const float* end_w = (const float*)d_in[24];
    const float* end_b = (const float*)d_in[25];

    int N = in_sizes[0] / 2;
    int E = in_sizes[1] / 7;
    int R = in_sizes[23];
    const int* src = ei;
    const int* dst = ei + E;

    float* ws = (float*)d_ws;
    size_t n64 = (size_t)N * 64;
    float* agg   = ws;
    float* zb    = ws + n64;
    float* bufA  = ws + 2 * n64;
    float* bufB  = ws + 3 * n64;
    float* stats = ws + 4 * n64;
    float* sums  = stats;
    float* sumsq = stats + 64;
    float* scale = stats + 128;
    float* shift = stats + 192;

    int rowTiles = (N + 15) / 16;
    int gblocks  = (rowTiles + 7) / 8;

    // -------- layer 1 (F=2)
    zero_k<<<(int)(((size_t)N * 2 + 255) / 256), 256, 0, stream>>>(agg, (size_t)N * 2);
    zero_k<<<1, 256, 0, stream>>>(sums, 128);
    edge_l1<<<(E + 255) / 256, 256, 0, stream>>>(x, ea, src, dst, el1_w, el1_b, agg, E);
    gemm1_l1<<<gblocks, 256, 0, stream>>>(x, agg, eps1, n1_w1, n1_b1, zb, sums, sumsq, N);
    bn_fin<<<1, 64, 0, stream>>>(sums, sumsq, n1_g, n1_bt, scale, shift, N);
    gemm2_h<<<gblocks, 256, 0, stream>>>(zb, scale, shift, n1_w2, n1_b2, bufA, N);

    // -------- layers 2 and 3 (shared weights)
    const float* cur = bufA;
    float* nxt = bufB;
    for (int l = 0; l < 2; ++l) {
        zero_k<<<(int)((n64 + 255) / 256), 256, 0, stream>>>(agg, n64);
        zero_k<<<1, 256, 0, stream>>>(sums, 128);
        edge_lH<<<(E + 7) / 8, 256, 0, stream>>>(cur, ea, src, dst, el2_w, el2_b, agg, E);
        gemm1_h<<<gblocks, 256, 0, stream>>>(cur, agg, eps2, n2_w1, n2_b1, zb, sums, sumsq, N);
        bn_fin<<<1, 64, 0, stream>>>(sums, sumsq, n2_g, n2_bt, scale, shift, N);
        gemm2_h<<<gblocks, 256, 0, stream>>>(zb, scale, shift, n2_w2, n2_b2, nxt, N);
        float* t = (float*)cur; cur = nxt; nxt = t;
    }

    // -------- fused regression head -> [N,1]
    reg_end<<<gblocks, 256, 0, stream>>>(cur, reg_w, reg_b, end_w, end_b,
                                         (float*)d_out, N, R);
}